// Attn_24051816858127
// MI455X (gfx1250) — compile-verified
//
#include <hip/hip_runtime.h>
#include <hip/hip_bf16.h>
#include <math.h>

typedef __attribute__((ext_vector_type(2))) float v2f;
typedef __attribute__((ext_vector_type(8))) float v8f;

#define D_DIM 1024
#define B_DIM 32
#define L_DIM 2048

// ---------------------------------------------------------------------------
// Kernel 1: v[b,d] = sum_e hidden[b,e] * W[e,d]
// M=32(b), N=1024(d), K=1024(e).  V_WMMA_F32_16X16X4_F32, one wave per
// 16x16 output tile, 256 K-steps of 4.
// A frag (16x4 f32): lanes 0-15 hold {K=0,K=1}, lanes 16-31 hold {K=2,K=3},
//                    M = lane&15 (ISA 7.12.2).
// B frag (4x16 f32): symmetric — lanes 0-15 rows K=0,1; lanes 16-31 rows
//                    K=2,3; N = lane&15.
// C/D  (16x16 f32):  VGPR r: lanes 0-15 -> M=r, lanes 16-31 -> M=r+8.
// ---------------------------------------------------------------------------
__global__ __launch_bounds__(128) void hw_gemm_wmma(const float* __restrict__ hidden,
                                                    const float* __restrict__ W,
                                                    float* __restrict__ v) {
    const int waveId = blockIdx.x * 4 + (threadIdx.x >> 5);
    const int lane   = threadIdx.x & 31;
    const int tileM  = waveId >> 6;   // 0..1   (M tiles: 32/16)
    const int tileN  = waveId & 63;   // 0..63  (N tiles: 1024/16)
    const int half   = lane >> 4;     // 0|1
    const int sub    = lane & 15;

    const int rowA = tileM * 16 + sub;   // hidden row (b)
    const int colB = tileN * 16 + sub;   // W column (d)

    v8f acc = {0.f, 0.f, 0.f, 0.f, 0.f, 0.f, 0.f, 0.f};
    for (int k0 = 0; k0 < D_DIM; k0 += 4) {
        const int ka = k0 + half * 2;
        v2f a, b;
        a.x = hidden[rowA * D_DIM + ka];
        a.y = hidden[rowA * D_DIM + ka + 1];
        b.x = W[(size_t)ka       * D_DIM + colB];
        b.y = W[(size_t)(ka + 1) * D_DIM + colB];
        acc = __builtin_amdgcn_wmma_f32_16x16x4_f32(
            /*neg_a=*/false, a, /*neg_b=*/false, b,
            /*c_mod=*/(short)0, acc, /*reuse_a=*/false, /*reuse_b=*/false);
    }

    const int mBase = tileM * 16 + half * 8;
    const int n     = tileN * 16 + sub;
#pragma unroll
    for (int r = 0; r < 8; ++r)
        v[(size_t)(mBase + r) * D_DIM + n] = acc[r];
}

// ---------------------------------------------------------------------------
// Wave reductions (wave32)
// ---------------------------------------------------------------------------
__device__ __forceinline__ float waveSum(float x) {
#pragma unroll
    for (int o = 16; o > 0; o >>= 1) x += __shfl_down(x, o, 32);
    return x;
}
__device__ __forceinline__ float waveMax(float x) {
#pragma unroll
    for (int o = 16; o > 0; o >>= 1) x = fmaxf(x, __shfl_down(x, o, 32));
    return x;
}

// ---------------------------------------------------------------------------
// Kernel 2: scores[b,l] = enc[l,b,:] . v[b,:]
// One wave per (l,b) pair; lane does 8x float4 of the 4KB contiguous row.
// Consecutive waves in a block read contiguous enc rows -> streaming 256MB
// at HBM rate; v (128KB) stays in L2. This is the roofline-limiting kernel.
// ---------------------------------------------------------------------------
__global__ __launch_bounds__(256) void scores_kernel(const float* __restrict__ enc,
                                                     const float* __restrict__ v,
                                                     float* __restrict__ scores) {
    const int wave = blockIdx.x * 8 + (threadIdx.x >> 5);
    const int lane = threadIdx.x & 31;
    const int l = wave >> 5;   // B_DIM == 32
    const int b = wave & 31;

    const float* __restrict__ e  = enc + ((size_t)l * B_DIM + b) * D_DIM;
    const float* __restrict__ vb = v + (size_t)b * D_DIM;

    float acc = 0.f;
#pragma unroll
    for (int i = 0; i < 8; ++i) {
        const int idx = i * 128 + lane * 4;            // contiguous 512B/step
        const float4 ev = *(const float4*)(e + idx);
        const float4 vv = *(const float4*)(vb + idx);
        acc = fmaf(ev.x, vv.x, acc);
        acc = fmaf(ev.y, vv.y, acc);
        acc = fmaf(ev.z, vv.z, acc);
        acc = fmaf(ev.w, vv.w, acc);
    }
    acc = waveSum(acc);
    if (lane == 0) scores[(size_t)b * L_DIM + l] = acc;
}

// ---------------------------------------------------------------------------
// Kernel 3: row softmax over l (2048) for each b (32 rows).
// Note: the reference's bias term contributes a constant per (b) row and
// cancels under softmax shift-invariance, so it is intentionally dropped.
// ---------------------------------------------------------------------------
__global__ __launch_bounds__(256) void softmax_rows(const float* __restrict__ scores,
                                                    float* __restrict__ out) {
    __shared__ float red[8];
    __shared__ float bcast;
    const int b    = blockIdx.x;
    const int tid  = threadIdx.x;
    const int lane = tid & 31;
    const int wid  = tid >> 5;
    const float* __restrict__ row = scores + (size_t)b * L_DIM;
    float* __restrict__ orow      = out    + (size_t)b * L_DIM;

    float vals[8];
    float m = -INFINITY;
#pragma unroll
    for (int i = 0; i < 8; ++i) {
        vals[i] = row[i * 256 + tid];
        m = fmaxf(m, vals[i]);
    }
    m = waveMax(m);
    if (lane == 0) red[wid] = m;
    __syncthreads();
    if (wid == 0) {
        float t = (lane < 8) ? red[lane] : -INFINITY;
        t = waveMax(t);
        if (lane == 0) bcast = t;
    }
    __syncthreads();
    m = bcast;

    float s = 0.f;
#pragma unroll
    for (int i = 0; i < 8; ++i) {
        vals[i] = __expf(vals[i] - m);
        s += vals[i];
    }
    s = waveSum(s);
    __syncthreads();
    if (lane == 0) red[wid] = s;
    __syncthreads();
    if (wid == 0) {
        float t = (lane < 8) ? red[lane] : 0.f;
        t = waveSum(t);
        if (lane == 0) bcast = t;
    }
    __syncthreads();

    const float inv = 1.0f / bcast;
#pragma unroll
    for (int i = 0; i < 8; ++i)
        orow[i * 256 + tid] = vals[i] * inv;
}

// ---------------------------------------------------------------------------
extern "C" void kernel_launch(void* const* d_in, const int* in_sizes, int n_in,
                              void* d_out, int out_size, void* d_ws, size_t ws_size,
                              hipStream_t stream) {
    const float* hidden = (const float*)d_in[0];   // [32, 1024]
    const float* enc    = (const float*)d_in[1];   // [2048, 32, 1024]
    const float* W      = (const float*)d_in[2];   // [1024, 1024]
    // d_in[3] (bias) is provably irrelevant to the softmax output (row-constant shift).
    float* out = (float*)d_out;                    // [32, 2048]

    float* v      = (float*)d_ws;                  // 32*1024 floats (128 KB)
    float* scores = v + B_DIM * D_DIM;             // 32*2048 floats (256 KB)

    // 128 output tiles (2 x 64), 4 waves/block -> 32 blocks
    hw_gemm_wmma<<<32, 128, 0, stream>>>(hidden, W, v);
    // 65536 (l,b) pairs, 8 waves/block -> 8192 blocks
    scores_kernel<<<(L_DIM * B_DIM) / 8, 256, 0, stream>>>(enc, v, scores);
    softmax_rows<<<B_DIM, 256, 0, stream>>>(scores, out);
}